// FastBEV_87445534146721
// MI455X (gfx1250) — compile-verified
//
#include <hip/hip_runtime.h>

// ---------------- problem constants (from reference) ----------------
constexpr int FH   = 225;
constexpr int FW   = 400;
constexpr int NXv  = 200;
constexpr int NYv  = 200;
constexpr int NZv  = 12;
constexpr int BB   = 2;
constexpr int NCAM = 6;
constexpr int CC   = 64;
constexpr int NPTS  = NXv * NYv * NZv;   // 480000
constexpr int PLANE = FH * FW;           // 90000 floats per (cam,channel) image
constexpr int OUTCH = CC * NZv;          // 768
constexpr int OUTPL = NYv * NXv;         // 40000

typedef float v2f __attribute__((ext_vector_type(2)));
typedef float v8f __attribute__((ext_vector_type(8)));

// ---------------------------------------------------------------------------
// Setup: fold  aug * (K*scale) * inv(Rc) * [l2e | tl - tc]  into one 3x4
// matrix P[b][cam] stored in workspace (2*6*12 = 144 floats).
// ---------------------------------------------------------------------------
__global__ void fastbev_setup(const float* __restrict__ l2e,
                              const float* __restrict__ c2e,
                              const float* __restrict__ intr,
                              const float* __restrict__ aug,
                              float* __restrict__ P)
{
    int t = threadIdx.x;
    if (t >= BB * NCAM) return;
    int b = t / NCAM, cam = t % NCAM;

    const float* L  = l2e  + b * 16;                  // [4][4]
    const float* Cm = c2e  + (b * NCAM + cam) * 16;   // [4][4]
    const float* K  = intr + (b * NCAM + cam) * 9;    // [3][3]
    const float* A  = aug  + (b * NCAM + cam) * 16;   // [4][4]

    // inv(Rc) via adjugate (mimics jnp.linalg.inv on the 3x3 rotation)
    float r00=Cm[0], r01=Cm[1], r02=Cm[2];
    float r10=Cm[4], r11=Cm[5], r12=Cm[6];
    float r20=Cm[8], r21=Cm[9], r22=Cm[10];
    float det = r00*(r11*r22-r12*r21) - r01*(r10*r22-r12*r20) + r02*(r10*r21-r11*r20);
    float id  = 1.0f / det;
    float I[3][3] = {
        {(r11*r22-r12*r21)*id, (r02*r21-r01*r22)*id, (r01*r12-r02*r11)*id},
        {(r12*r20-r10*r22)*id, (r00*r22-r02*r20)*id, (r02*r10-r00*r12)*id},
        {(r10*r21-r11*r20)*id, (r01*r20-r00*r21)*id, (r00*r11-r01*r10)*id}};

    // K scaled by stride on rows 0,1
    float Ks[3][3] = {{K[0]*0.25f, K[1]*0.25f, K[2]*0.25f},
                      {K[3]*0.25f, K[4]*0.25f, K[5]*0.25f},
                      {K[6],       K[7],       K[8]      }};

    // M = Ks * inv(Rc)
    float M[3][3];
    for (int r = 0; r < 3; ++r)
        for (int c = 0; c < 3; ++c)
            M[r][c] = Ks[r][0]*I[0][c] + Ks[r][1]*I[1][c] + Ks[r][2]*I[2][c];

    // R = Ra * M
    float R[3][3];
    for (int r = 0; r < 3; ++r)
        for (int c = 0; c < 3; ++c)
            R[r][c] = A[r*4+0]*M[0][c] + A[r*4+1]*M[1][c] + A[r*4+2]*M[2][c];

    // Rtot = R * Rl ;  ttot = R * (tl - tc) + ta
    float d0 = L[3]  - Cm[3];
    float d1 = L[7]  - Cm[7];
    float d2 = L[11] - Cm[11];
    float* Pm = P + (b * NCAM + cam) * 12;            // [3][4]
    for (int r = 0; r < 3; ++r) {
        Pm[r*4+0] = R[r][0]*L[0] + R[r][1]*L[4] + R[r][2]*L[8];
        Pm[r*4+1] = R[r][0]*L[1] + R[r][1]*L[5] + R[r][2]*L[9];
        Pm[r*4+2] = R[r][0]*L[2] + R[r][1]*L[6] + R[r][2]*L[10];
        Pm[r*4+3] = R[r][0]*d0   + R[r][1]*d1   + R[r][2]*d2 + A[r*4+3];
    }
}

// ---------------------------------------------------------------------------
// Main kernel: one thread per voxel, enumerated in OUTPUT order
// (ix fastest -> coalesced stores). Projection of 32 points x 6 cams is done
// with 4x V_WMMA_F32_16X16X4_F32 per wave, transposed back via LDS.
// Output stores use non-temporal hints so the 246 MB write stream does not
// evict the (L2-resident, reused) 138 MB/batch feature maps.
// ---------------------------------------------------------------------------
__global__ __launch_bounds__(256) void fastbev_main(const float* __restrict__ feat,
                                                    const float* __restrict__ P,
                                                    float* __restrict__ out)
{
    __shared__ float sh[8][32][18];    // per-wave: 32 points x (6 cams * {u,v,w})

    const int tid  = threadIdx.x;
    const int wave = tid >> 5;
    const int lane = tid & 31;
    const int b    = blockIdx.y;
    const int gblk = blockIdx.x * 256;        // block's base voxel (output order)
    const int gwav = gblk + wave * 32;        // wave's base voxel

    const int N  = lane & 15;                 // WMMA column / point-local index
    const int hi = lane >> 4;                 // low/high half of the wave

    // ---- build B operands for the two cam-groups (constant per wave) ----
    // B[K][N] = P[cam(N)][row(N)][K] ; VGPR0 holds K={0|2}, VGPR1 holds K={1|3}
    v2f bmat[2];
    #pragma unroll
    for (int h = 0; h < 2; ++h) {
        float b0 = 0.f, b1 = 0.f;
        if (N < 9) {
            int cam = 3 * h + N / 3;
            int row = N - (N / 3) * 3;
            const float* Pm = P + ((b * NCAM + cam) * 3 + row) * 4;
            int K0 = hi ? 2 : 0;
            b0 = Pm[K0];
            b1 = Pm[K0 + 1];
        }
        bmat[h] = (v2f){b0, b1};
    }

    // ---- 2 point-groups x 2 cam-groups = 4 WMMAs ----
    #pragma unroll
    for (int grp = 0; grp < 2; ++grp) {
        // A[M][K] = homogeneous coords of point gwav + grp*16 + M
        // VGPR0 = K={0|2} (x|z), VGPR1 = K={1|3} (y|1)
        int gA = gwav + grp * 16 + N;
        int ixA = gA % NXv;
        int rA  = gA / NXv;
        int iyA = rA % NYv;
        int izA = rA / NYv;
        float a0, a1;
        if (hi == 0) { a0 = (float)ixA * 0.5f - 50.0f; a1 = (float)iyA * 0.5f - 50.0f; }
        else         { a0 = (float)izA * 0.5f - 3.0f;  a1 = 1.0f; }
        v2f amat = (v2f){a0, a1};

        #pragma unroll
        for (int h = 0; h < 2; ++h) {
            v8f cz = {};
            v8f d = __builtin_amdgcn_wmma_f32_16x16x4_f32(
                /*neg_a=*/false, amat, /*neg_b=*/false, bmat[h],
                /*c_mod=*/(short)0, cz, /*reuse_a=*/false, /*reuse_b=*/false);
            if (N < 9) {
                #pragma unroll
                for (int r = 0; r < 8; ++r)
                    sh[wave][grp * 16 + 8 * hi + r][h * 9 + N] = d[r];
            }
        }
    }

    __syncthreads();

    // ---- per-lane: pick last valid camera, then gather/scatter 64 channels ----
    const int g  = gwav + lane;
    const int ix = g % NXv;
    const int rr = g / NXv;
    const int iy = rr % NYv;
    const int iz = rr / NYv;

    int sel_cam = -1;
    int sel_idx = 0;
    #pragma unroll
    for (int cam = 0; cam < NCAM; ++cam) {
        float u = sh[wave][lane][3 * cam + 0];
        float v = sh[wave][lane][3 * cam + 1];
        float w = sh[wave][lane][3 * cam + 2];
        float xi = rintf(u / w);   // round-half-even, matches jnp.round
        float yi = rintf(v / w);
        bool valid = (w > 0.0f) && (xi >= 0.0f) && (yi >= 0.0f)
                   && (xi < (float)FW) && (yi < (float)FH);
        if (valid) { sel_cam = cam; sel_idx = (int)yi * FW + (int)xi; }
    }

    const bool  has  = (sel_cam >= 0);
    const float mask = has ? 1.0f : 0.0f;   // computed once; select applied per load
    const int   cam  = has ? sel_cam : 0;
    const float* fp = feat + ((b * NCAM + cam) * CC) * PLANE + sel_idx;

    // out[((b*768 + c*12 + iz) * 200 + iy) * 200 + ix]
    float* op = out + (long)b * OUTCH * OUTPL + iz * OUTPL + iy * NXv + ix;

    #pragma unroll 16
    for (int c = 0; c < CC; ++c) {
        float val = fp[c * PLANE];
        val = has ? val : 0.0f;
        __builtin_nontemporal_store(val, op + c * (NZv * OUTPL));
    }
    (void)mask;
}

// ---------------------------------------------------------------------------
extern "C" void kernel_launch(void* const* d_in, const int* in_sizes, int n_in,
                              void* d_out, int out_size, void* d_ws, size_t ws_size,
                              hipStream_t stream)
{
    (void)in_sizes; (void)n_in; (void)out_size; (void)ws_size;
    const float* feat = (const float*)d_in[0];
    // d_in[1] = points (recomputed analytically; construction is exact in fp32)
    const float* l2e  = (const float*)d_in[2];
    const float* c2e  = (const float*)d_in[3];
    const float* intr = (const float*)d_in[4];
    const float* aug  = (const float*)d_in[5];
    float* P = (float*)d_ws;                 // 144 floats of combined matrices

    fastbev_setup<<<1, 16, 0, stream>>>(l2e, c2e, intr, aug, P);

    dim3 grid(NPTS / 256, BB);               // 1875 x 2, exact cover, EXEC all-1s
    fastbev_main<<<grid, 256, 0, stream>>>(feat, P, (float*)d_out);
}